// DynamicTargetLoss_7773890805999
// MI455X (gfx1250) — compile-verified
//
#include <hip/hip_runtime.h>
#include <hip/hip_bf16.h>
#include <climits>

// ---------------------------------------------------------------------------
// DynamicTargetLoss for MI455X (gfx1250, wave32).
//   B=1024, T=16384 f32. loss = first prefix-mean of column sums of
//   (pred-target)^2 exceeding 1.9, else the last prefix-mean (index T-2).
// Pass 1 is HBM-bound (128 MiB @ 23.3 TB/s ~ 5.5us). The over-B reduction is
// done with V_WMMA_F32_16X16X4_F32 using an all-ones A matrix:
//   D = ones(16x4) x SQ(4x16) + C  ->  every row of D = column sums of tile.
// ---------------------------------------------------------------------------

#define B_DIM 1024
#define T_DIM 16384
#define ROW_CHUNKS 8                     // grid.y row split
#define ROWS_PER_CHUNK (B_DIM / ROW_CHUNKS)   // 128
#define COLS_PER_WAVE 16                 // WMMA N
#define WAVES_PER_BLOCK 8
#define COLS_PER_BLOCK (COLS_PER_WAVE * WAVES_PER_BLOCK) // 128
#define TARGET_LOSS 1.9

typedef __attribute__((ext_vector_type(2))) float v2f;
typedef __attribute__((ext_vector_type(8))) float v8f;

// Pass 1: per (rowChunk, 16-column tile) wave computes
//   partial[rowChunk*T + col] = sum_{r in chunk} (p[r][col]-t[r][col])^2
// via accumulating WMMA with ones-A. B-matrix per ISA layout:
//   lane: n = lane&15 (column), kpair = lane>>4 -> rows {r+2*kpair, r+2*kpair+1}.
__global__ void __launch_bounds__(256)
colsum_sq_wmma_kernel(const float* __restrict__ p, const float* __restrict__ t,
                      float* __restrict__ partial) {
    const int lane = threadIdx.x & 31;
    const int wave = threadIdx.x >> 5;
    const int n    = lane & 15;
    const int kb   = (lane >> 4) << 1;           // 0 or 2
    const int col0 = blockIdx.x * COLS_PER_BLOCK + wave * COLS_PER_WAVE;
    const int r0   = blockIdx.y * ROWS_PER_CHUNK;

    size_t base = (size_t)(r0 + kb) * T_DIM + (size_t)(col0 + n);
    const float* pp = p + base;
    const float* tt = t + base;

    v2f ones; ones.x = 1.0f; ones.y = 1.0f;      // A = 16x4 all-ones
    v8f acc = {};                                 // C/D 16x16 f32 accumulator

    #pragma unroll 4
    for (int it = 0; it < ROWS_PER_CHUNK / 4; ++it) {
        float p0 = __builtin_nontemporal_load(pp);
        float p1 = __builtin_nontemporal_load(pp + T_DIM);
        float t0 = __builtin_nontemporal_load(tt);
        float t1 = __builtin_nontemporal_load(tt + T_DIM);
        float d0 = p0 - t0;
        float d1 = p1 - t1;
        v2f b; b.x = d0 * d0; b.y = d1 * d1;     // B = 4x16 squared diffs
        // (neg_a, A, neg_b, B, c_mod, C, reuse_a, reuse_b)
        acc = __builtin_amdgcn_wmma_f32_16x16x4_f32(
            false, ones, false, b, (short)0, acc, false, false);
        pp += (size_t)4 * T_DIM;
        tt += (size_t)4 * T_DIM;
    }

    // D row 0 (VGPR0, lanes 0-15) holds the 16 column sums for this tile.
    if (lane < 16) {
        partial[(size_t)blockIdx.y * T_DIM + (size_t)(col0 + lane)] = acc[0];
    }
}

// Pass 2 (single block, 256 threads): reduce 8 partials per column, prefix-scan
// in double, find first prefix-mean > TARGET_LOSS over c in [0, T-2].
__global__ void __launch_bounds__(256)
finalize_kernel(const float* __restrict__ partial, float* __restrict__ out) {
    const int tid = threadIdx.x;
    __shared__ double   s_sum[256];
    __shared__ double   s_pre[256];
    __shared__ unsigned s_minidx;
    __shared__ float    s_val;
    __shared__ float    s_fb;
    if (tid == 0) s_minidx = 0xFFFFFFFFu;

    const int c0 = tid * (T_DIM / 256);          // 64 contiguous columns/thread

    // local chunk sum (col sums re-read from L2 in phase 2; 512 KB is L2-hot)
    double local = 0.0;
    for (int j = 0; j < T_DIM / 256; ++j) {
        const int c = c0 + j;
        float s = 0.0f;
        #pragma unroll
        for (int ch = 0; ch < ROW_CHUNKS; ++ch)
            s += partial[(size_t)ch * T_DIM + c];
        local += (double)s;
    }
    s_sum[tid] = local;
    __syncthreads();

    if (tid == 0) {                              // serial 256-entry exclusive scan
        double run = 0.0;
        for (int i = 0; i < 256; ++i) { s_pre[i] = run; run += s_sum[i]; }
    }
    __syncthreads();

    double run = s_pre[tid];
    int my = INT_MAX;
    double myMean = 0.0;
    for (int j = 0; j < T_DIM / 256; ++j) {
        const int c = c0 + j;
        float s = 0.0f;
        #pragma unroll
        for (int ch = 0; ch < ROW_CHUNKS; ++ch)
            s += partial[(size_t)ch * T_DIM + c];
        run += (double)s;
        if (c <= T_DIM - 2) {
            const double mean = run / ((double)B_DIM * (double)(c + 1));
            if (c == T_DIM - 2) s_fb = (float)mean;               // fallback mean
            if (my == INT_MAX && mean > TARGET_LOSS) { my = c; myMean = mean; }
        }
    }
    atomicMin(&s_minidx, (unsigned)my);
    __syncthreads();
    if (my != INT_MAX && (unsigned)my == s_minidx) s_val = (float)myMean;
    __syncthreads();
    if (tid == 0) out[0] = (s_minidx < (unsigned)T_DIM) ? s_val : s_fb;
}

extern "C" void kernel_launch(void* const* d_in, const int* in_sizes, int n_in,
                              void* d_out, int out_size, void* d_ws, size_t ws_size,
                              hipStream_t stream) {
    (void)in_sizes; (void)n_in; (void)out_size; (void)ws_size;
    const float* pred = (const float*)d_in[0];
    const float* targ = (const float*)d_in[1];
    float* out = (float*)d_out;
    float* partial = (float*)d_ws;               // ROW_CHUNKS * T_DIM floats = 512 KB

    dim3 grid(T_DIM / COLS_PER_BLOCK, ROW_CHUNKS);   // (128, 8)
    colsum_sq_wmma_kernel<<<grid, 256, 0, stream>>>(pred, targ, partial);
    finalize_kernel<<<1, 256, 0, stream>>>(partial, out);
}